// KymatioPhaseScattering1D_27805618274435
// MI455X (gfx1250) — compile-verified
//
#include <hip/hip_runtime.h>
#include <math.h>
#include <stdint.h>

// Problem constants (from the reference): J=6, Q=4, N=8192, NPAD=16384, N_FILT=24
#define NFILT  24
#define NBATCH 8
#define NSIG   8192
#define NPADC  16384
#define PADL   4096
#define LSTR   129           // LDS row stride (banks: gcd(129,64)=1)
#define MATF   (128 * LSTR)  // floats per 128x128 LDS matrix component

typedef __attribute__((ext_vector_type(2))) float v2f;
typedef __attribute__((ext_vector_type(8))) float v8f;
typedef __attribute__((ext_vector_type(4))) unsigned int v4u;
typedef __attribute__((ext_vector_type(4))) int v4i;
typedef __attribute__((ext_vector_type(8))) int v8i;

__device__ __forceinline__ int ldsidx(int r, int c) { return r * LSTR + c; }

// numpy 'reflect' (no edge duplication) index into [0, NSIG)
__device__ __forceinline__ int reflect_idx(int t) {
  const int period = 2 * NSIG - 2;
  t %= period;
  if (t < 0) t += period;
  if (t >= NSIG) t = period - t;
  return t;
}

#define WMMA_F32(A, B, C) \
  __builtin_amdgcn_wmma_f32_16x16x4_f32(false, (A), false, (B), (short)0, (C), false, false)

// ---------------------------------------------------------------------------
// TDM: DMA one row of 16384 dwords from global into LDS with 1 dword of
// padding every 128 dwords  ->  element e lands at (e>>7)*129 + (e&127),
// i.e. exactly the ldsidx() layout. D# built per CDNA5 ISA 8.3/8.4.
// Issue from a single wave; completion via TENSORcnt.
// ---------------------------------------------------------------------------
__device__ __forceinline__ void tdm_load_row_padded(const float* gsrc,
                                                    unsigned lds_byte_off) {
  const unsigned long long ga = (unsigned long long)(uintptr_t)gsrc;
  v4u g0;
  g0[0] = 1u;                                   // count=1 (valid user D#)
  g0[1] = lds_byte_off;                         // lds_addr (bytes)
  g0[2] = (unsigned)(ga & 0xFFFFFFFFu);         // global_addr[31:0]
  g0[3] = (unsigned)((ga >> 32) & 0x1FFFFFFu)   // global_addr[56:32]
        | (2u << 30);                           // type = 2 ("image")
  v8i g1;
  g1[0] = (int)((2u << 16)        // data_size = 2 -> 4 bytes
              | (1u << 20)        // pad_enable
              | (6u << 22));      // pad_interval = 6 -> 128 dwords
                                  // pad_amount  = 0 -> 1 dword
  g1[1] = (int)(16384u << 16);    // tensor_dim0[15:0]=16384 (bits 79:64... low half)
  g1[2] = (int)(1u << 16);        // tensor_dim0 hi=0 ; tensor_dim1 = 1
  g1[3] = (int)(16384u << 16);    // tensor_dim1 hi=0 ; tile_dim0 = 16384
  g1[4] = 0;                      // tile_dim1 = 0 (1-D), tile_dim2 = 0
  g1[5] = 16384;                  // tensor_dim0_stride (low 32)
  g1[6] = 0;
  g1[7] = 0;
  const v4i z4 = {0, 0, 0, 0};
#if __has_include(<hip/amd_detail/amd_gfx1250_TDM.h>)
  const v8i z8 = {0, 0, 0, 0, 0, 0, 0, 0};
  __builtin_amdgcn_tensor_load_to_lds(g0, g1, z4, z4, z8, 0);
#else
  __builtin_amdgcn_tensor_load_to_lds(g0, g1, z4, z4, 0);
#endif
}

// ---------------------------------------------------------------------------
// D = F * S : 128x128 complex matmul. F (DFT matrix, symmetric) in global,
// row-major stride 128, imaginary part scaled by `sign`; S in LDS.
// Bn: B[k][n] = S[ldsidx(k,n)]  (normal);  Bt: B[k][n] = S[ldsidx(n,k)].
// 256 threads = 8 waves; wave w owns output rows [16w,16w+16).
// A fragment: one float2 (b64) load pair per k-step, reused across 4 tiles.
// ---------------------------------------------------------------------------
template <bool BT>
__device__ __forceinline__ void mm_F_S(const float* __restrict__ Fre,
                                       const float* __restrict__ Fim, float sign,
                                       const float* Sre, const float* Sim,
                                       float* Dre, float* Dim) {
  const int lane = threadIdx.x & 31;
  const int wave = threadIdx.x >> 5;
  const int m0 = wave * 16;
  const int lm = lane & 15;       // row (A) / col (B,D) within tile
  const int g  = lane >> 4;       // half-wave group
  const int kg = g * 2;           // K sub-offset for A/B fragments
  const float2* __restrict__ Fre2 = (const float2*)Fre;
  const float2* __restrict__ Fim2 = (const float2*)Fim;

  for (int half = 0; half < 2; ++half) {
    v8f accre[4], accim[4];
#pragma unroll
    for (int t = 0; t < 4; ++t) { accre[t] = (v8f){}; accim[t] = (v8f){}; }

    for (int k = 0; k < 128; k += 4) {
      const int fa = (m0 + lm) * 128 + (k + kg);   // even -> 8B aligned
      const float2 ar2 = Fre2[fa >> 1];
      const float2 ai2 = Fim2[fa >> 1];
      v2f are, aim;
      are.x = ar2.x;        are.y = ar2.y;
      aim.x = sign * ai2.x; aim.y = sign * ai2.y;
      v2f ain = -aim;
#pragma unroll
      for (int t = 0; t < 4; ++t) {
        const int n0 = (half * 4 + t) * 16;
        v2f bre, bim;
        if (BT) {
          const int fb = ldsidx(n0 + lm, k + kg);
          bre.x = Sre[fb]; bre.y = Sre[fb + 1];
          bim.x = Sim[fb]; bim.y = Sim[fb + 1];
        } else {
          const int fb = ldsidx(k + kg, n0 + lm);
          bre.x = Sre[fb]; bre.y = Sre[fb + LSTR];
          bim.x = Sim[fb]; bim.y = Sim[fb + LSTR];
        }
        accre[t] = WMMA_F32(are, bre, accre[t]);
        accre[t] = WMMA_F32(ain, bim, accre[t]);
        accim[t] = WMMA_F32(are, bim, accim[t]);
        accim[t] = WMMA_F32(aim, bre, accim[t]);
      }
    }
#pragma unroll
    for (int t = 0; t < 4; ++t) {
      const int n0 = (half * 4 + t) * 16;
#pragma unroll
      for (int i = 0; i < 8; ++i) {    // D layout: vgpr i -> M = i + 8*g
        const int r = m0 + i + 8 * g;
        const int c = n0 + lm;
        Dre[ldsidx(r, c)] = accre[t][i];
        Dim[ldsidx(r, c)] = accim[t][i];
      }
    }
  }
}

// ---------------------------------------------------------------------------
// 16384-point FFT, four-step with symmetric-F reformulation:
//   b0 holds the *linear* fill: b0[ldsidx(e>>7, e&127)] = x[e]  (= A^T)
//   P = F * A^T        (P[k2][n1] = inner[n1][k2])          -> b1
//   P *= TW(r*c)       (symmetric table, linear indexing)
//   C = F * P^T        (C[k1][k2]; bin k at ldsidx(k>>7,k&127)) -> b0
// sign=+1 forward, -1 inverse (no 1/N scale).
// ---------------------------------------------------------------------------
__device__ __forceinline__ void fft16k(float* b0re, float* b0im,
                                       float* b1re, float* b1im,
                                       const float* Fre, const float* Fim,
                                       const float* TWre, const float* TWim,
                                       float sign) {
  __syncthreads();
  mm_F_S<false>(Fre, Fim, sign, b0re, b0im, b1re, b1im);   // P = F * A^T
  __syncthreads();
  for (int e = threadIdx.x; e < NPADC; e += 256) {         // twiddle W_N^(r*c)
    const int li = ldsidx(e >> 7, e & 127);
    const float tr = TWre[e], ti = sign * TWim[e];
    const float xr = b1re[li], xi = b1im[li];
    b1re[li] = xr * tr - xi * ti;
    b1im[li] = xr * ti + xi * tr;
  }
  __syncthreads();
  mm_F_S<true>(Fre, Fim, sign, b1re, b1im, b0re, b0im);    // C = F * P^T
  __syncthreads();
}

// ---------------------------------------------------------------------------
// Kernel 0: build DFT-128 matrix and NPAD twiddles into workspace
// ---------------------------------------------------------------------------
__global__ void k_tables(float* Fre, float* Fim, float* TWre, float* TWim) {
  const int i = blockIdx.x * blockDim.x + threadIdx.x;
  if (i >= 128 * 128) return;
  const int r = i >> 7, c = i & 127;
  const double TWO_PI = 6.283185307179586476925286766559;
  const double af = -TWO_PI * (double)(r * c) / 128.0;
  Fre[i] = (float)cos(af);
  Fim[i] = (float)sin(af);
  const double at = -TWO_PI * (double)(r * c) / 16384.0;
  TWre[i] = (float)cos(at);
  TWim[i] = (float)sin(at);
}

// ---------------------------------------------------------------------------
// Kernel 1: reflect-pad x, forward FFT -> xf (8 x 16384 complex)
// ---------------------------------------------------------------------------
__global__ void __launch_bounds__(256)
k_fft_x(const float* __restrict__ x,
        const float* __restrict__ Fre, const float* __restrict__ Fim,
        const float* __restrict__ TWre, const float* __restrict__ TWim,
        float* __restrict__ xfre, float* __restrict__ xfim) {
  extern __shared__ float lds[];
  float* b0re = lds;
  float* b0im = lds + MATF;
  float* b1re = lds + 2 * MATF;
  float* b1im = lds + 3 * MATF;
  const int b = blockIdx.x;

  for (int e = threadIdx.x; e < NPADC; e += 256) {   // linear fill (coalesced)
    const int src = reflect_idx(e - PADL);
    const int li = ldsidx(e >> 7, e & 127);
    b0re[li] = x[b * NSIG + src];
    b0im[li] = 0.0f;
  }
  fft16k(b0re, b0im, b1re, b1im, Fre, Fim, TWre, TWim, 1.0f);
  for (int e = threadIdx.x; e < NPADC; e += 256) {
    const int li = ldsidx(e >> 7, e & 127);
    xfre[b * NPADC + e] = b0re[li];
    xfim[b * NPADC + e] = b0im[li];
  }
}

// ---------------------------------------------------------------------------
// Kernel 2: filt[b][f] = crop(ifft(xf[b] * psi[f]))  (192 blocks)
// xf row is DMA'd into the padded LDS layout by the Tensor Data Mover.
// ---------------------------------------------------------------------------
__global__ void __launch_bounds__(256)
k_filt(const float* __restrict__ xfre, const float* __restrict__ xfim,
       const float* __restrict__ psi,
       const float* __restrict__ Fre, const float* __restrict__ Fim,
       const float* __restrict__ TWre, const float* __restrict__ TWim,
       float* __restrict__ fre, float* __restrict__ fim) {
  extern __shared__ float lds[];
  float* b0re = lds;
  float* b0im = lds + MATF;
  float* b1re = lds + 2 * MATF;
  float* b1im = lds + 3 * MATF;
  const int b = blockIdx.x / NFILT;
  const int f = blockIdx.x % NFILT;

  // TDM load of both spectrum components into the stride-129 layout
  if (threadIdx.x < 32) {   // wave 0 only (TDM is a wave-level DMA op)
    tdm_load_row_padded(xfre + (size_t)b * NPADC, 0u);
    tdm_load_row_padded(xfim + (size_t)b * NPADC,
                        (unsigned)(MATF * sizeof(float)));
    __builtin_amdgcn_s_wait_tensorcnt(0);
  }
  __syncthreads();

  for (int e = threadIdx.x; e < NPADC; e += 256) {   // apply psi in place
    const float pr = psi[f * NPADC + e];
    const int li = ldsidx(e >> 7, e & 127);
    b0re[li] *= pr;
    b0im[li] *= pr;
  }
  fft16k(b0re, b0im, b1re, b1im, Fre, Fim, TWre, TWim, -1.0f);  // inverse
  const float sc = 1.0f / (float)NPADC;
  float* dre = fre + (size_t)(b * NFILT + f) * NSIG;
  float* dim = fim + (size_t)(b * NFILT + f) * NSIG;
  for (int t = threadIdx.x; t < NSIG; t += 256) {
    const int n = PADL + t;
    const int li = ldsidx(n >> 7, n & 127);
    dre[t] = b0re[li] * sc;
    dim[t] = b0im[li] * sc;
  }
}

// ---------------------------------------------------------------------------
// Kernel 3: per (batch, pair): phase-harmonic correlation, reflect-pad,
// forward FFT, low-pass phi on first 256 bins, 256-pt IDFT, 128 real taps.
// ---------------------------------------------------------------------------
__global__ void __launch_bounds__(256)
k_pairs(const float* __restrict__ fre, const float* __restrict__ fim,
        const float* __restrict__ phi, const float* __restrict__ powers,
        const int* __restrict__ i_idx, const int* __restrict__ j_idx,
        const float* __restrict__ Fre, const float* __restrict__ Fim,
        const float* __restrict__ TWre, const float* __restrict__ TWim,
        float* __restrict__ out, int np) {
  extern __shared__ float lds[];
  float* b0re = lds;
  float* b0im = lds + MATF;
  float* b1re = lds + 2 * MATF;
  float* b1im = lds + 3 * MATF;
  const int b = blockIdx.x / np;
  const int p = blockIdx.x % np;
  const float pw = powers[p];
  const float* sire = fre + (size_t)(b * NFILT + i_idx[p]) * NSIG;
  const float* siim = fim + (size_t)(b * NFILT + i_idx[p]) * NSIG;
  const float* sjre = fre + (size_t)(b * NFILT + j_idx[p]) * NSIG;
  const float* sjim = fim + (size_t)(b * NFILT + j_idx[p]) * NSIG;

  // Warm the L2/WGP$ path for the four streamed rows (global_prefetch_b8)
  {
    const int t = threadIdx.x * 128;   // one line every 128 B across 32 KB head
    if (t < NSIG * (int)sizeof(float)) {
      __builtin_prefetch((const char*)sire + t, 0, 1);
      __builtin_prefetch((const char*)siim + t, 0, 1);
      __builtin_prefetch((const char*)sjre + t, 0, 1);
      __builtin_prefetch((const char*)sjim + t, 0, 1);
    }
  }

  for (int e = threadIdx.x; e < NPADC; e += 256) {   // linear fill (coalesced)
    const int s = reflect_idx(e - PADL);
    const float sr = sire[s], si = siim[s];
    const float mag = sqrtf(sr * sr + si * si);
    const float ph = atan2f(si, sr) * pw;
    float sp, cp;
    __sincosf(ph, &sp, &cp);
    const float ar = mag * cp, ai = mag * sp;
    const float jr = sjre[s], ji = sjim[s];
    const int li = ldsidx(e >> 7, e & 127);
    b0re[li] = ar * jr + ai * ji;    // acc * conj(sj)
    b0im[li] = ai * jr - ar * ji;
  }
  fft16k(b0re, b0im, b1re, b1im, Fre, Fim, TWre, TWim, 1.0f);

  // sm = ifft( (X .* phi)[0:256] ); out = Re(sm[64:192])
  if (threadIdx.x < 128) {
    const int m = 64 + threadIdx.x;
    float acc = 0.0f;
    const float w = 6.28318530717958647692f / 256.0f;
    for (int k = 0; k < 256; ++k) {
      const int li = ldsidx(k >> 7, k & 127);
      const float ph_k = phi[k];
      const float xr = b0re[li] * ph_k;
      const float xi = b0im[li] * ph_k;
      float s, c;
      __sincosf(w * (float)(m * k), &s, &c);
      acc += xr * c - xi * s;
    }
    out[((size_t)b * np + p) * 128 + threadIdx.x] = acc * (1.0f / 256.0f);
  }
}

// ---------------------------------------------------------------------------
extern "C" void kernel_launch(void* const* d_in, const int* in_sizes, int n_in,
                              void* d_out, int out_size, void* d_ws, size_t ws_size,
                              hipStream_t stream) {
  const float* x      = (const float*)d_in[0];   // (8, 8192)
  const float* psi    = (const float*)d_in[1];   // (24, 16384)
  const float* phi    = (const float*)d_in[2];   // (16384,)
  const float* powers = (const float*)d_in[3];   // (300,)
  const int*   i_idx  = (const int*)d_in[4];
  const int*   j_idx  = (const int*)d_in[5];
  const int np = in_sizes[3];                    // 300 pairs

  // Workspace layout (floats)
  float* ws   = (float*)d_ws;
  float* Fre  = ws;                    // 16384
  float* Fim  = Fre + 128 * 128;       // 16384
  float* TWre = Fim + 128 * 128;       // 16384
  float* TWim = TWre + 128 * 128;      // 16384
  float* xfre = TWim + 128 * 128;      // 8*16384
  float* xfim = xfre + NBATCH * NPADC;
  float* fre  = xfim + NBATCH * NPADC; // 8*24*8192
  float* fim  = fre + (size_t)NBATCH * NFILT * NSIG;

  const size_t shmem = (size_t)4 * MATF * sizeof(float);  // 258 KB < 320 KB/WGP

  k_tables<<<64, 256, 0, stream>>>(Fre, Fim, TWre, TWim);
  k_fft_x<<<NBATCH, 256, shmem, stream>>>(x, Fre, Fim, TWre, TWim, xfre, xfim);
  k_filt<<<NBATCH * NFILT, 256, shmem, stream>>>(xfre, xfim, psi,
                                                 Fre, Fim, TWre, TWim, fre, fim);
  k_pairs<<<NBATCH * np, 256, shmem, stream>>>(fre, fim, phi, powers, i_idx, j_idx,
                                               Fre, Fim, TWre, TWim,
                                               (float*)d_out, np);
}